// Cheb_GCNN_10L_uw_54485955117439
// MI455X (gfx1250) — compile-verified
//
#include <hip/hip_runtime.h>
#include <hip/hip_bf16.h>

#define NN  50000
#define EE  1200000
#define INF 128
#define CC  64
#define BN_EPS 1e-5f

typedef float v2f __attribute__((ext_vector_type(2)));
typedef float v8f __attribute__((ext_vector_type(8)));

// ---------------- utility ----------------
__global__ __launch_bounds__(256) void zero_kernel(float* __restrict__ p, long long n) {
    long long i = (long long)blockIdx.x * blockDim.x + threadIdx.x;
    long long s = (long long)gridDim.x * blockDim.x;
    for (; i < n; i += s) p[i] = 0.0f;
}

// ---------------- graph normalization ----------------
__global__ __launch_bounds__(256) void degree_kernel(const int* __restrict__ row,
                                                     float* __restrict__ deg) {
    int e = blockIdx.x * 256 + threadIdx.x;
    if (e < EE) atomicAdd(&deg[row[e]], 1.0f);
}

__global__ __launch_bounds__(256) void dinv_kernel(float* __restrict__ deg) {
    int i = blockIdx.x * 256 + threadIdx.x;
    if (i < NN) { float d = deg[i]; deg[i] = (d > 0.0f) ? rsqrtf(d) : 0.0f; }
}

__global__ __launch_bounds__(256) void edgew_kernel(const int* __restrict__ row,
                                                    const int* __restrict__ col,
                                                    const float* __restrict__ dinv,
                                                    float* __restrict__ w) {
    int e = blockIdx.x * 256 + threadIdx.x;
    if (e < EE) w[e] = -dinv[row[e]] * dinv[col[e]];
}

// ---------------- effective weights: [W0 - W2 | W1 | 2*W2] ----------------
// Hoists the Chebyshev recombination out of the GEMM hot loop.
__global__ __launch_bounds__(256) void weff_kernel(const float* __restrict__ W, // 3 x F x 64
                                                   float* __restrict__ Weff,    // 3 x F x 64
                                                   int FC) {                    // F*64
    int i = blockIdx.x * 256 + threadIdx.x;
    if (i < FC) {
        float w0 = W[i], w1 = W[FC + i], w2 = W[2 * FC + i];
        Weff[i]          = w0 - w2;
        Weff[FC + i]     = w1;
        Weff[2 * FC + i] = w2 + w2;
    }
}

// ---------------- SpMM: dst[row[e], f] += w[e] * src[col[e], f] ----------------
// 4 channels per thread: float4 gather (global_load_b128) + 4 coalesced f32
// atomics; edge index/weight loads amortized 4x. Channels map to consecutive
// lanes so row/col/w are uniform within groups of F/4 lanes.
__global__ __launch_bounds__(256) void spmm_kernel(const int* __restrict__ row,
                                                   const int* __restrict__ col,
                                                   const float* __restrict__ w,
                                                   const float* __restrict__ src,
                                                   float* __restrict__ dst,
                                                   int logQ, int F) {          // logQ = log2(F/4)
    long long tid = (long long)blockIdx.x * 256 + threadIdx.x;
    long long total = (long long)EE << logQ;
    if (tid >= total) return;
    int e  = (int)(tid >> logQ);
    int f0 = (int)(tid & ((1 << logQ) - 1)) * 4;
    int r = row[e], c = col[e];
    float we = w[e];
    const float4 v = *(const float4*)(src + (size_t)c * F + f0);
    float* d = dst + (size_t)r * F + f0;
    atomicAdd(d + 0, we * v.x);
    atomicAdd(d + 1, we * v.y);
    atomicAdd(d + 2, we * v.z);
    atomicAdd(d + 3, we * v.w);
}

// ---------------- fused 3-way GEMM + bias + ReLU + BN partial sums ----------------
// out = A0*Weff0 + T1*Weff1 + Z*Weff2 + b ; act = relu(out); accumulate sum/sumsq.
// One wave computes a 16x64 tile via v_wmma_f32_16x16x4_f32 (full fp32 path).
__global__ __launch_bounds__(256) void cheb_gemm_bn(const float* __restrict__ A0,
                                                    const float* __restrict__ T1,
                                                    const float* __restrict__ Z,
                                                    const float* __restrict__ Weff, // 3 x F x 64
                                                    const float* __restrict__ bias,
                                                    float* __restrict__ act,        // N x 64
                                                    float* __restrict__ bnsum,      // 64
                                                    float* __restrict__ bnsq,       // 64
                                                    int F) {
    __shared__ float s_sum[CC];
    __shared__ float s_sq[CC];
    const int lane = threadIdx.x & 31;
    const int wave = threadIdx.x >> 5;
    const int half = lane >> 4;     // 0: lanes 0-15, 1: lanes 16-31
    const int l15  = lane & 15;
    if (threadIdx.x < CC) { s_sum[threadIdx.x] = 0.0f; s_sq[threadIdx.x] = 0.0f; }
    __syncthreads();

    const int tile = blockIdx.x * 8 + wave;          // wave-uniform
    if (tile * 16 < NN) {                            // uniform branch: EXEC stays all-1s
        v8f acc[4] = {};
        const size_t rowBase = (size_t)(tile * 16 + l15) * F;
        for (int src = 0; src < 3; ++src) {
            const float* __restrict__ A  = (src == 0) ? A0 : (src == 1) ? T1 : Z;
            const float* __restrict__ Ws = Weff + (size_t)src * F * CC;
            for (int k0 = 0; k0 < F; k0 += 4) {
                const int kx = k0 + 2 * half;        // K pair handled by this half-wave
                v2f a;                                // A frag: M=l15, K=kx,kx+1 (b64 load)
                a.x = A[rowBase + kx];
                a.y = A[rowBase + kx + 1];
                const float* __restrict__ wr0 = Ws + (size_t)kx * CC;
                const float* __restrict__ wr1 = wr0 + CC;
#pragma unroll
                for (int t = 0; t < 4; ++t) {        // 4 N-tiles of 16 channels
                    const int n = t * 16 + l15;
                    v2f b; b.x = wr0[n]; b.y = wr1[n]; // B frag: K=kx,kx+1, N=n
                    acc[t] = __builtin_amdgcn_wmma_f32_16x16x4_f32(
                        false, a, false, b, (short)0, acc[t], false, false);
                }
            }
        }
        // epilogue: +bias, relu, store, per-channel partial sums
#pragma unroll
        for (int t = 0; t < 4; ++t) {
            const int n  = t * 16 + l15;
            const float bv = bias[n];
            float s1 = 0.0f, s2 = 0.0f;
#pragma unroll
            for (int r = 0; r < 8; ++r) {
                const int m = r + 8 * half;           // C/D layout: M = r + 8*half
                float v = acc[t][r] + bv;
                v = (v > 0.0f) ? v : 0.0f;
                act[(size_t)(tile * 16 + m) * CC + n] = v;
                s1 += v; s2 += v * v;
            }
            atomicAdd(&s_sum[n], s1);
            atomicAdd(&s_sq[n],  s2);
        }
    }
    __syncthreads();
    if (threadIdx.x < CC) {
        atomicAdd(&bnsum[threadIdx.x], s_sum[threadIdx.x]);
        atomicAdd(&bnsq[threadIdx.x],  s_sq[threadIdx.x]);
    }
}

// ---------------- BN finalize + apply ----------------
__global__ void bn_finalize(const float* __restrict__ bnsum,
                            const float* __restrict__ bnsq,
                            float* __restrict__ musig) {
    int c = threadIdx.x;
    if (c < CC) {
        float mu  = bnsum[c] * (1.0f / NN);
        float var = bnsq[c] * (1.0f / NN) - mu * mu;
        musig[c]      = mu;
        musig[CC + c] = rsqrtf(var + BN_EPS);
    }
}

__global__ __launch_bounds__(256) void bn_apply(const float* __restrict__ act,
                                                const float* __restrict__ musig,
                                                const float* __restrict__ gamma,
                                                const float* __restrict__ beta,
                                                float* __restrict__ h) {
    long long i = (long long)blockIdx.x * 256 + threadIdx.x;
    if (i < (long long)NN * CC) {
        int c = (int)(i & (CC - 1));
        h[i] = gamma[c] * (act[i] - musig[c]) * musig[CC + c] + beta[c];
    }
}

// ---------------- final linear C=64 -> 2 ----------------
__global__ __launch_bounds__(256) void linear_kernel(const float* __restrict__ h,
                                                     const float* __restrict__ lw, // 64x2
                                                     const float* __restrict__ lb, // 2
                                                     float* __restrict__ out) {
    int n = blockIdx.x * 256 + threadIdx.x;
    if (n < NN) {
        const float* hr = h + (size_t)n * CC;
        float o0 = lb[0], o1 = lb[1];
#pragma unroll 8
        for (int k = 0; k < CC; ++k) {
            float v = hr[k];
            o0 += v * lw[k * 2 + 0];
            o1 += v * lw[k * 2 + 1];
        }
        out[n * 2 + 0] = o0;
        out[n * 2 + 1] = o1;
    }
}

extern "C" void kernel_launch(void* const* d_in, const int* in_sizes, int n_in,
                              void* d_out, int out_size, void* d_ws, size_t ws_size,
                              hipStream_t stream) {
    const float* x     = (const float*)d_in[0];
    const int*   ei    = (const int*)d_in[1];
    const float* W1    = (const float*)d_in[2];
    const float* b1    = (const float*)d_in[3];
    const float* Wr    = (const float*)d_in[4];
    const float* br    = (const float*)d_in[5];
    const float* gamma = (const float*)d_in[6];
    const float* beta  = (const float*)d_in[7];
    const float* lw    = (const float*)d_in[8];
    const float* lb    = (const float*)d_in[9];
    float* out = (float*)d_out;

    const int* row = ei;
    const int* col = ei + EE;

    float* ws    = (float*)d_ws;
    float* deg   = ws;                            // N      (becomes dinv in-place)
    float* ew    = deg + NN;                      // E
    float* y1    = ew + EE;                       // N*128  (Tx1)
    float* y2    = y1 + (size_t)NN * INF;         // N*128  (Z = Lhat Tx1)
    float* act   = y2 + (size_t)NN * INF;         // N*64
    float* hA    = act + (size_t)NN * CC;         // N*64
    float* hB    = hA + (size_t)NN * CC;          // N*64
    float* bnacc = hB + (size_t)NN * CC;          // 128 (sum | sumsq)
    float* musig = bnacc + 2 * CC;                // 128 (mu | rsigma)
    float* weff  = musig + 2 * CC;                // 3*128*64 max

    // graph normalization: deg -> dinv -> edge weights of L_hat
    zero_kernel<<<256, 256, 0, stream>>>(deg, NN);
    degree_kernel<<<(EE + 255) / 256, 256, 0, stream>>>(row, deg);
    dinv_kernel<<<(NN + 255) / 256, 256, 0, stream>>>(deg);
    edgew_kernel<<<(EE + 255) / 256, 256, 0, stream>>>(row, col, deg, ew);

    const float* cur = x;
    float* nxt = hA;
    for (int li = 0; li < 10; ++li) {
        const int F    = (li == 0) ? INF : CC;
        const int logQ = (li == 0) ? 5 : 4;       // log2(F/4)
        const float* Wl = (li == 0) ? W1 : Wr + (size_t)(li - 1) * 3 * CC * CC;
        const float* bl = (li == 0) ? b1 : br + (size_t)(li - 1) * CC;
        const long long nF = (long long)NN * F;
        const long long eQ = (long long)EE << logQ;
        const int FC = F * CC;

        weff_kernel<<<(FC + 255) / 256, 256, 0, stream>>>(Wl, weff, FC);

        zero_kernel<<<2048, 256, 0, stream>>>(y1, nF);
        spmm_kernel<<<(unsigned)((eQ + 255) / 256), 256, 0, stream>>>(row, col, ew, cur, y1, logQ, F);
        zero_kernel<<<2048, 256, 0, stream>>>(y2, nF);
        spmm_kernel<<<(unsigned)((eQ + 255) / 256), 256, 0, stream>>>(row, col, ew, y1, y2, logQ, F);
        zero_kernel<<<1, 256, 0, stream>>>(bnacc, 2 * CC);

        cheb_gemm_bn<<<(NN / 16 + 7) / 8, 256, 0, stream>>>(cur, y1, y2, weff, bl,
                                                            act, bnacc, bnacc + CC, F);
        bn_finalize<<<1, CC, 0, stream>>>(bnacc, bnacc + CC, musig);
        bn_apply<<<(unsigned)(((long long)NN * CC + 255) / 256), 256, 0, stream>>>(
            act, musig, gamma + (size_t)li * CC, beta + (size_t)li * CC, nxt);

        cur = nxt;
        nxt = (nxt == hA) ? hB : hA;
    }
    linear_kernel<<<(NN + 255) / 256, 256, 0, stream>>>(cur, lw, lb, out);
}